// EdgeConvTorch_19842748908334
// MI455X (gfx1250) — compile-verified
//
#include <hip/hip_runtime.h>
#include <cfloat>

typedef __attribute__((ext_vector_type(2))) float v2f;
typedef __attribute__((ext_vector_type(8))) float v8f;

#define BATCH  16
#define CIN    64
#define COUT   64
#define NPTS   2048
#define KNN    20
#define EPSV   1e-5f
#define LDSW   2052            // padded row pitch (dwords) to dodge bank conflicts
#define CNT    (BATCH * NPTS * KNN)

// ---------------------------------------------------------------------------
// Kernel 1: squared norms  xx[b,n] = sum_c x[b,c,n]^2
// ---------------------------------------------------------------------------
__global__ void norms_kernel(const float* __restrict__ x, float* __restrict__ xx) {
    int p = blockIdx.x * blockDim.x + threadIdx.x;      // p = b*N + n
    if (p >= BATCH * NPTS) return;
    int b = p >> 11, n = p & (NPTS - 1);
    const float* xb = x + (size_t)b * CIN * NPTS + n;
    float s = 0.f;
#pragma unroll
    for (int c = 0; c < CIN; ++c) { float v = xb[c * NPTS]; s += v * v; }
    xx[p] = s;
}

// ---------------------------------------------------------------------------
// Kernel 2: fused Gram (f32 WMMA) -> pairwise-dist slab in LDS -> top-K
// grid: BATCH * (NPTS/16) blocks of 256 threads (8 waves).
// Each block owns 16 query rows; LDS slab = 16 x 2052 floats (~128 KB, CDNA5
// 320KB/WGP). Wave w computes column tiles m = w, w+8, ... via 16 chained
// V_WMMA_F32_16X16X4_F32, then waves cooperatively extract top-20 per row.
// ---------------------------------------------------------------------------
__global__ void knn_kernel(const float* __restrict__ x, const float* __restrict__ xx,
                           int* __restrict__ knn_idx) {
    extern __shared__ float pdsm[];                     // [16][LDSW]
    const int b   = blockIdx.x >> 7;
    const int r0  = (blockIdx.x & 127) << 4;
    const int wid  = threadIdx.x >> 5;
    const int lane = threadIdx.x & 31;
    const int half = lane >> 4;                         // 0: lanes 0-15, 1: 16-31
    const int lrow = lane & 15;
    const float* xb = x + (size_t)b * CIN * NPTS;

    // Row norms for this lane's 8 output rows (C/D layout: row = v + 8*half)
    float xxr[8];
#pragma unroll
    for (int v = 0; v < 8; ++v) xxr[v] = xx[b * NPTS + r0 + v + 8 * half];

    // Preload A fragments (16x64 query tile), invariant across column tiles.
    // A layout (16x4 f32): lanes 0-15 rows M=0..15 w/ K={4kb,4kb+1}; lanes
    // 16-31 same rows w/ K={4kb+2,4kb+3}.
    v2f afrag[16];
#pragma unroll
    for (int kb = 0; kb < 16; ++kb) {
        int k0 = 4 * kb + 2 * half;
        afrag[kb].x = xb[(size_t)k0 * NPTS + r0 + lrow];
        afrag[kb].y = xb[(size_t)(k0 + 1) * NPTS + r0 + lrow];
    }

    // 16 column tiles per wave, strided by 8 waves
    for (int tt = 0; tt < 16; ++tt) {
        const int m  = wid + 8 * tt;                    // column tile 0..127
        const int nc = m * 16 + lrow;
        v8f acc = {};
#pragma unroll
        for (int kb = 0; kb < 16; ++kb) {
            int k0 = 4 * kb + 2 * half;
            v2f bfrag;
            bfrag.x = xb[(size_t)k0 * NPTS + nc];
            bfrag.y = xb[(size_t)(k0 + 1) * NPTS + nc];
            acc = __builtin_amdgcn_wmma_f32_16x16x4_f32(
                false, afrag[kb], false, bfrag, (short)0, acc, false, false);
        }
        // pd = 2*g - ||row||^2 - ||col||^2  -> LDS slab
        float xxc = xx[b * NPTS + nc];
#pragma unroll
        for (int v = 0; v < 8; ++v) {
            int row = v + 8 * half;
            pdsm[row * LDSW + nc] = 2.f * acc[v] - xxr[v] - xxc;
        }
    }
    __syncthreads();

    // Top-20 per row: wave w handles rows w and w+8. 20 rounds of strided
    // scan + shfl-xor argmax butterfly; owner lane blanks the winner.
    for (int rr = 0; rr < 2; ++rr) {
        const int row = wid + 8 * rr;
        float* prow = pdsm + row * LDSW;
        int* orow = knn_idx + ((size_t)(b * NPTS) + r0 + row) * KNN;
        for (int t = 0; t < KNN; ++t) {
            float bval = -FLT_MAX; int bidx = 0x7fffffff;
            for (int cc = lane; cc < NPTS; cc += 32) {
                float v = prow[cc];
                if (v > bval) { bval = v; bidx = cc; }
            }
#pragma unroll
            for (int off = 16; off > 0; off >>= 1) {
                float ov = __shfl_xor(bval, off, 32);
                int   oi = __shfl_xor(bidx, off, 32);
                if (ov > bval || (ov == bval && oi < bidx)) { bval = ov; bidx = oi; }
            }
            if (lane == (bidx & 31)) prow[bidx] = -FLT_MAX;   // owner blanks it
            if (lane == 0) orow[t] = bidx;
        }
    }
}

// ---------------------------------------------------------------------------
// Kernel 3: y1[b,n,o] = (W1-W2)·x[:,n] + bias[o],  y2[b,n,o] = W2·x[:,n]
// (b,n,c) layout so the K gathers later read contiguous 64-float vectors.
// grid: BATCH * (NPTS/16) blocks of 128 threads (4 waves); wave w owns the
// 16-wide output-channel tile [16w,16w+16). Two WMMA accumulators share B.
// ---------------------------------------------------------------------------
__global__ void feat_kernel(const float* __restrict__ x, const float* __restrict__ W,
                            const float* __restrict__ bias,
                            float* __restrict__ y1, float* __restrict__ y2) {
    const int b   = blockIdx.x >> 7;
    const int n0  = (blockIdx.x & 127) << 4;
    const int wid  = threadIdx.x >> 5;
    const int lane = threadIdx.x & 31;
    const int half = lane >> 4;
    const int lrow = lane & 15;
    const int ob = wid * 16;
    const float* xb = x + (size_t)b * CIN * NPTS;
    const int n = n0 + lrow;

    v8f acc1 = {}, acc2 = {};
#pragma unroll
    for (int kb = 0; kb < 16; ++kb) {
        int k0 = 4 * kb + 2 * half;
        int o  = ob + lrow;
        v2f a1, a2, bf;
        float w2x = W[o * 128 + 64 + k0], w2y = W[o * 128 + 64 + k0 + 1];
        a1.x = W[o * 128 + k0] - w2x;  a1.y = W[o * 128 + k0 + 1] - w2y;
        a2.x = w2x;                    a2.y = w2y;
        bf.x = xb[(size_t)k0 * NPTS + n];
        bf.y = xb[(size_t)(k0 + 1) * NPTS + n];
        acc1 = __builtin_amdgcn_wmma_f32_16x16x4_f32(false, a1, false, bf, (short)0, acc1, false, false);
        acc2 = __builtin_amdgcn_wmma_f32_16x16x4_f32(false, a2, false, bf, (short)0, acc2, false, false);
    }
    float* y1p = y1 + ((size_t)(b * NPTS) + n) * COUT;
    float* y2p = y2 + ((size_t)(b * NPTS) + n) * COUT;
#pragma unroll
    for (int v = 0; v < 8; ++v) {
        int o = ob + v + 8 * half;
        y1p[o] = acc1[v] + bias[o];    // fold bias into y1
        y2p[o] = acc2[v];
    }
}

// ---------------------------------------------------------------------------
// Kernel 4a/4b: BN statistics (sum, sumsq per channel) over h = y1 + y2[idx]
// ---------------------------------------------------------------------------
__global__ void zero_stats_kernel(float* __restrict__ stats) {
    if (threadIdx.x < 128) stats[threadIdx.x] = 0.f;
}

__global__ void stats_kernel(const float* __restrict__ y1, const float* __restrict__ y2,
                             const int* __restrict__ idx, float* __restrict__ stats) {
    __shared__ float ssum[COUT], ssum2[COUT];
    int tid = threadIdx.x;
    if (tid < COUT) { ssum[tid] = 0.f; ssum2[tid] = 0.f; }
    __syncthreads();
    int c = tid & 63, g = tid >> 6;                     // 4 (b,n) workers/block
    float s = 0.f, s2 = 0.f;
    for (int p = blockIdx.x * 4 + g; p < BATCH * NPTS; p += gridDim.x * 4) {
        int b = p >> 11;
        float y1v = y1[(size_t)p * COUT + c];
        const int* ip = idx + (size_t)p * KNN;
        const float* y2b = y2 + (size_t)b * NPTS * COUT;
#pragma unroll
        for (int t = 0; t < KNN; ++t) {
            float h = y1v + y2b[(size_t)ip[t] * COUT + c];
            s += h; s2 += h * h;
        }
    }
    atomicAdd(&ssum[c], s); atomicAdd(&ssum2[c], s2);
    __syncthreads();
    if (tid < COUT) { atomicAdd(&stats[tid], ssum[tid]); atomicAdd(&stats[COUT + tid], ssum2[tid]); }
}

// ---------------------------------------------------------------------------
// Kernel 5: out[b,o,n] = max(0, max_k (scale*h + shift))
// ---------------------------------------------------------------------------
__global__ void out_kernel(const float* __restrict__ y1, const float* __restrict__ y2,
                           const int* __restrict__ idx, const float* __restrict__ stats,
                           const float* __restrict__ gamma, const float* __restrict__ beta,
                           float* __restrict__ out) {
    int tid = threadIdx.x;
    int c = tid & 63, g = tid >> 6;
    int p = blockIdx.x * 4 + g;                         // p = b*N + n
    if (p >= BATCH * NPTS) return;
    int b = p >> 11, n = p & (NPTS - 1);
    const float inv_cnt = 1.f / (float)CNT;
    float mean  = stats[c] * inv_cnt;
    float var   = stats[COUT + c] * inv_cnt - mean * mean;
    float scale = gamma[c] * rsqrtf(var + EPSV);
    float shift = beta[c] - mean * scale;
    float y1v = y1[(size_t)p * COUT + c];
    const int* ip = idx + (size_t)p * KNN;
    const float* y2b = y2 + (size_t)b * NPTS * COUT;
    float m = 0.f;                                      // relu floor
#pragma unroll
    for (int t = 0; t < KNN; ++t) {
        float h = y1v + y2b[(size_t)ip[t] * COUT + c];
        m = fmaxf(m, scale * h + shift);
    }
    out[((size_t)b * COUT + c) * NPTS + n] = m;
}

// ---------------------------------------------------------------------------
extern "C" void kernel_launch(void* const* d_in, const int* in_sizes, int n_in,
                              void* d_out, int out_size, void* d_ws, size_t ws_size,
                              hipStream_t stream) {
    const float* x     = (const float*)d_in[0];
    const float* W     = (const float*)d_in[1];
    const float* bias  = (const float*)d_in[2];
    const float* gamma = (const float*)d_in[3];
    const float* beta  = (const float*)d_in[4];
    float* out = (float*)d_out;

    // workspace layout
    char* ws = (char*)d_ws;
    float* xx   = (float*)ws;                              // 32768 f
    int*   idx  = (int*)(ws + 131072);                     // 655360 i32
    float* y1   = (float*)(ws + 131072 + 2621440);         // 2M f
    float* y2   = y1 + (size_t)BATCH * NPTS * COUT;        // 2M f
    float* stats = y2 + (size_t)BATCH * NPTS * COUT;       // 128 f

    norms_kernel<<<(BATCH * NPTS) / 256, 256, 0, stream>>>(x, xx);

    size_t knn_lds = (size_t)16 * LDSW * sizeof(float);    // ~128 KB (CDNA5 WGP LDS)
    knn_kernel<<<BATCH * (NPTS / 16), 256, knn_lds, stream>>>(x, xx, idx);

    feat_kernel<<<BATCH * (NPTS / 16), 128, 0, stream>>>(x, W, bias, y1, y2);

    zero_stats_kernel<<<1, 128, 0, stream>>>(stats);
    stats_kernel<<<512, 256, 0, stream>>>(y1, y2, idx, stats);

    out_kernel<<<(BATCH * NPTS) / 4, 256, 0, stream>>>(y1, y2, idx, stats, gamma, beta, out);
}